// ComputeLBP_full_53_55585466745022
// MI455X (gfx1250) — compile-verified
//
#include <hip/hip_runtime.h>
#include <hip/hip_bf16.h>
#include <stdint.h>

typedef __attribute__((ext_vector_type(2))) float v2f;
typedef __attribute__((ext_vector_type(8))) float v8f;

// Uniform LBP bin indices (58 entries), from the reference UNIS table.
__constant__ int c_unis[58] = {
    0, 1, 2, 3, 4, 6, 7, 8, 12, 14, 15, 16, 24, 28, 30, 31, 32, 48, 56, 60,
    62, 63, 64, 96, 112, 120, 124, 126, 127, 128, 129, 131, 135, 143, 159,
    191, 192, 193, 195, 199, 207, 223, 224, 225, 227, 231, 239, 240, 241,
    243, 247, 248, 249, 251, 252, 253, 254, 255};

__global__ void zero_hist_kernel(unsigned* __restrict__ hist, int n) {
  int i = blockIdx.x * blockDim.x + threadIdx.x;
  if (i < n) hist[i] = 0u;
}

__device__ __forceinline__ unsigned quant(float f) {
  // matches (x*255.0).astype(uint8): truncation toward zero, x in [0,1)
  return (unsigned)(f * 255.0f);
}

// Load nquads aligned float4's of row `row` starting at col `cstart`
// (multiple of 4, may be negative), quantize to u8 range, zero outside image.
template <int NQ>
__device__ __forceinline__ void load_quant_row(const float* __restrict__ img,
                                               int row, int cstart,
                                               unsigned* __restrict__ q) {
  if ((unsigned)row < 512u) {
    const float* rp = img + row * 512;
#pragma unroll
    for (int i = 0; i < NQ; ++i) {
      const int cq = cstart + i * 4;
      if ((unsigned)cq < 512u) {  // aligned quad: fully in or fully out
        const float4 v = *(const float4*)(rp + cq);
        q[i * 4 + 0] = quant(v.x);
        q[i * 4 + 1] = quant(v.y);
        q[i * 4 + 2] = quant(v.z);
        q[i * 4 + 3] = quant(v.w);
      } else {
        q[i * 4 + 0] = 0u;
        q[i * 4 + 1] = 0u;
        q[i * 4 + 2] = 0u;
        q[i * 4 + 3] = 0u;
      }
    }
  } else {
#pragma unroll
    for (int i = 0; i < NQ * 4; ++i) q[i] = 0u;
  }
}

// LBP-5x3 code + per-batch histogram.
// Grid: (64 row-chunks, B batches). Block: 256 threads = 8 waves.
// Wave w owns row r0+w; each lane owns 16 consecutive pixels (col = lane*16).
// Per lane: 20 global_load_b128 feed 16 pixels via register-window reuse.
__global__ __launch_bounds__(256) void lbp_hist_kernel(
    const float* __restrict__ x, unsigned* __restrict__ hist) {
  __shared__ unsigned sh[256];
  const int t = threadIdx.x;
  sh[t] = 0u;
  __syncthreads();

  const int batch = blockIdx.y;
  const int r = (blockIdx.x << 3) + (t >> 5);  // this wave's row
  const int c = (t & 31) << 4;                 // lane's 16-pixel segment base
  const float* img = x + (size_t)batch * (512 * 512);

  // Center row r: cols c-8..c+23  (need c-5..c+20)  -> 8 quads
  // Rows r-3 / r+3: cols c-4..c+19 (need c-3..c+18) -> 6 quads each
  unsigned qc[32], qu[24], qd[24];
  load_quant_row<8>(img, r, c - 8, qc);
  load_quant_row<6>(img, r - 3, c - 4, qu);
  load_quant_row<6>(img, r + 3, c - 4, qd);

#pragma unroll
  for (int j = 0; j < 16; ++j) {
    // index maps: qc[idx] = col c-8+idx ; qu/qd[idx] = col c-4+idx
    const unsigned center = qc[j + 8];
    unsigned code = 0u;
    code |= (qu[j + 4]  >= center) ? 1u   : 0u;  // (-3,  0)
    code |= (qu[j + 7]  >= center) ? 2u   : 0u;  // (-3, +3)
    code |= (qc[j + 13] >= center) ? 4u   : 0u;  // ( 0, +5)
    code |= (qd[j + 7]  >= center) ? 8u   : 0u;  // (+3, +3)
    code |= (qd[j + 4]  >= center) ? 16u  : 0u;  // (+3,  0)
    code |= (qd[j + 1]  >= center) ? 32u  : 0u;  // (+3, -3)
    code |= (qc[j + 3]  >= center) ? 64u  : 0u;  // ( 0, -5)
    code |= (qu[j + 1]  >= center) ? 128u : 0u;  // (-3, -3)
    atomicAdd(&sh[code], 1u);
  }

  __syncthreads();
  const unsigned v = sh[t];
  if (v) atomicAdd(&hist[batch * 256 + t], v);
}

// out[B,59] = (hist[B,256] x S[256,64])[:, :59] mod 256, computed exactly with
// V_WMMA_F32_16X16X4_F32 (all values are integers < 2^24 -> f32 FMA is exact).
// S[k, map[k]] = 1, where map[k] = uniform index or 58 ("rest" column).
__global__ __launch_bounds__(256) void wmma_reduce_kernel(
    const unsigned* __restrict__ hist, float* __restrict__ out, int B) {
  __shared__ int smap[256];
  const int t = threadIdx.x;
  {
    int m = 58;
#pragma unroll
    for (int j = 0; j < 58; ++j)
      if (c_unis[j] == t) m = j;
    smap[t] = m;
  }
  __syncthreads();

  const int wave = t >> 5;
  const int lane = t & 31;
  const int lm = lane & 15;
  const int hi = (lane >> 4) & 1;   // lane half selects K pair (A and B)
  const int mtiles = B >> 4;        // 8 for B=128
  const int ntotal = mtiles * 4;    // N=64 -> 4 n-tiles

  for (int tile = wave; tile < ntotal; tile += 8) {
    const int mt = tile >> 2;
    const int nt = tile & 3;
    const int m = (mt << 4) + lm;   // A row (batch) for this lane
    const int n = (nt << 4) + lm;   // B/D column for this lane

    v8f acc = {};
    const unsigned* hrow = hist + (size_t)m * 256;
    for (int k0 = 0; k0 < 256; k0 += 4) {
      const int ka = k0 + (hi ? 2 : 0);
      // A: 16x4 f32, lane m holds K=ka, ka+1 in 2 VGPRs
      v2f A = {(float)hrow[ka], (float)hrow[ka + 1]};
      // B: 4x16 f32, lane n holds K=ka, ka+1 in 2 VGPRs
      v2f Bm = {(smap[ka] == n) ? 1.0f : 0.0f,
                (smap[ka + 1] == n) ? 1.0f : 0.0f};
      acc = __builtin_amdgcn_wmma_f32_16x16x4_f32(
          /*neg_a=*/false, A, /*neg_b=*/false, Bm,
          /*c_mod=*/(short)0, acc, /*reuse_a=*/false, /*reuse_b=*/false);
    }

    if (n < 59) {
      // D layout: VGPR r holds M = mt*16 + hi*8 + r, column n = lane%16
#pragma unroll
      for (int r = 0; r < 8; ++r) {
        const int M = (mt << 4) + (hi << 3) + r;
        const unsigned iv = (unsigned)(acc[r] + 0.5f);  // exact integer
        out[M * 59 + n] = (float)(iv & 255u);
      }
    }
  }
}

extern "C" void kernel_launch(void* const* d_in, const int* in_sizes, int n_in,
                              void* d_out, int out_size, void* d_ws,
                              size_t ws_size, hipStream_t stream) {
  const float* x = (const float*)d_in[0];
  float* out = (float*)d_out;
  unsigned* hist = (unsigned*)d_ws;  // B*256 uint32 (128 KB for B=128)

  const int B = in_sizes[0] >> 18;   // / (512*512)

  zero_hist_kernel<<<(B * 256 + 255) / 256, 256, 0, stream>>>(hist, B * 256);
  lbp_hist_kernel<<<dim3(64, B), 256, 0, stream>>>(x, hist);
  wmma_reduce_kernel<<<1, 256, 0, stream>>>(hist, out, B);
}